// Transformer_28329604284483
// MI455X (gfx1250) — compile-verified
//
#include <hip/hip_runtime.h>
#include <math.h>

typedef float v2f __attribute__((ext_vector_type(2)));
typedef float v8f __attribute__((ext_vector_type(8)));

#define D_MODEL 768
#define F_FF    3072
#define SEQ     512
#define BATCH   4
#define HEADS   12
#define HDIM    64
#define ROWS    (BATCH * SEQ)      // 2048
#define NBLOCKS 12

__device__ __forceinline__ v8f wmma_f32(v2f a, v2f b, v8f c) {
  // V_WMMA_F32_16X16X4_F32 : D = A(16x4) x B(4x16) + C(16x16), full fp32
  return __builtin_amdgcn_wmma_f32_16x16x4_f32(
      /*neg_a=*/false, a, /*neg_b=*/false, b,
      /*c_mod=*/(short)0, c, /*reuse_a=*/false, /*reuse_b=*/false);
}

// ---------------------------------------------------------------------------
// LayerNorm: one 256-thread block per row (D=768 -> 3 elems/thread)
// ---------------------------------------------------------------------------
__global__ __launch_bounds__(256)
void ln_kernel(const float* __restrict__ x, const float* __restrict__ g,
               const float* __restrict__ be, float* __restrict__ y) {
  __shared__ float red[256];
  const int row = blockIdx.x;
  const int t = threadIdx.x;
  const float* xr = x + (size_t)row * D_MODEL;

  float v0 = xr[t], v1 = xr[t + 256], v2 = xr[t + 512];

  red[t] = v0 + v1 + v2;
  __syncthreads();
  for (int off = 128; off > 0; off >>= 1) {
    if (t < off) red[t] += red[t + off];
    __syncthreads();
  }
  const float mu = red[0] * (1.0f / (float)D_MODEL);
  __syncthreads();

  const float d0 = v0 - mu, d1 = v1 - mu, d2 = v2 - mu;
  red[t] = d0 * d0 + d1 * d1 + d2 * d2;
  __syncthreads();
  for (int off = 128; off > 0; off >>= 1) {
    if (t < off) red[t] += red[t + off];
    __syncthreads();
  }
  const float var = red[0] * (1.0f / (float)D_MODEL);
  const float rs = rsqrtf(var + 1e-5f);

  float* yr = y + (size_t)row * D_MODEL;
  yr[t]       = d0 * rs * g[t]       + be[t];
  yr[t + 256] = d1 * rs * g[t + 256] + be[t + 256];
  yr[t + 512] = d2 * rs * g[t + 512] + be[t + 512];
}

// ---------------------------------------------------------------------------
// fp32 WMMA GEMM: C[M x Nn] = A[M x K] @ W[K x Nn] + bias (+GELU | +residual)
// 128x64 block tile, 8 waves of 32x32. Double-buffered LDS (global latency
// hidden under WMMAs) + paired-float2 B layout (single 8B LDS load per B
// fragment) + register-level fragment prefetch inside the k-loop.
// ---------------------------------------------------------------------------
#define BM 128
#define BN 64
#define BK 32
#define AS_STRIDE  36   // floats: conflict-free A frag reads, 16B-aligned fills
#define BSP_STRIDE 80   // float2: lanes16-31 land +32 banks -> conflict-free

#define OP_NONE 0
#define OP_GELU 1
#define OP_RES  2

__global__ __launch_bounds__(256)
void gemm_kernel(const float* __restrict__ A, const float* __restrict__ W,
                 const float* __restrict__ bias, float* __restrict__ C,
                 const float* __restrict__ R, int K, int Nn, int op) {
  __shared__ float  As[2][BM * AS_STRIDE];        // 2 x 18.0 KB
  __shared__ float2 Bs[2][(BK / 2) * BSP_STRIDE]; // 2 x 10.0 KB

  const int tid  = threadIdx.x;
  const int lane = tid & 31;
  const int wave = tid >> 5;
  const int wm = wave & 3;              // 4 wave-rows of 32
  const int wn = wave >> 2;             // 2 wave-cols of 32
  const int l16 = lane & 15;
  const int lhi = lane >> 4;            // 0|1
  const int khalf = lhi << 1;           // 0|2

  const int mBlock = blockIdx.x * BM;
  const int nBlock = blockIdx.y * BN;
  const int n0 = wn * 32 + l16;

  // staging registers (double buffer in regs, single barrier per stage)
  float4 ra[4];
  float4 rb0, rb1;
  const int akr = tid >> 3;             // A fill: row group base
  const int akc = (tid & 7) * 4;        //         col within 32
  const int bkp = tid >> 4;             // B fill: k-pair 0..15
  const int bkc = (tid & 15) * 4;       //         col 0..60

  auto gloadA = [&](int kk) {
#pragma unroll
    for (int i = 0; i < 4; i++) {
      int r = akr + i * 32;
      ra[i] = *(const float4*)(A + (size_t)(mBlock + r) * K + kk + akc);
    }
  };
  auto gloadB = [&](int kk) {
    const float* wp = W + (size_t)(kk + 2 * bkp) * Nn + nBlock + bkc;
    rb0 = *(const float4*)wp;
    rb1 = *(const float4*)(wp + Nn);
  };
  auto stage = [&](int buf) {
#pragma unroll
    for (int i = 0; i < 4; i++)
      *(float4*)&As[buf][(akr + i * 32) * AS_STRIDE + akc] = ra[i];
    float2* dst = &Bs[buf][bkp * BSP_STRIDE + bkc];
    dst[0] = make_float2(rb0.x, rb1.x);
    dst[1] = make_float2(rb0.y, rb1.y);
    dst[2] = make_float2(rb0.z, rb1.z);
    dst[3] = make_float2(rb0.w, rb1.w);
  };

  v8f acc[2][2] = {};

  gloadA(0);
  gloadB(0);
  int buf = 0;
  for (int kk = 0; kk < K; kk += BK) {
    stage(buf);
    __syncthreads();                       // includes s_wait_dscnt fence
    if (kk + BK < K) { gloadA(kk + BK); gloadB(kk + BK); }  // hide under WMMAs

    const float*  AsB = As[buf];
    const float2* BsB = Bs[buf];
    v2f a0c, a1c, b0c, b1c, a0n, a1n, b0n, b1n;

    auto ldfrag = [&](int k4, v2f& A0, v2f& A1, v2f& B0, v2f& B1) {
      const int kb = k4 * 4 + khalf;
      A0 = *(const v2f*)&AsB[(wm * 32 + l16) * AS_STRIDE + kb];
      A1 = *(const v2f*)&AsB[(wm * 32 + 16 + l16) * AS_STRIDE + kb];
      const int kp = k4 * 2 + lhi;
      B0 = *(const v2f*)&BsB[kp * BSP_STRIDE + n0];
      B1 = *(const v2f*)&BsB[kp * BSP_STRIDE + n0 + 16];
    };

    ldfrag(0, a0c, a1c, b0c, b1c);
#pragma unroll
    for (int k4 = 0; k4 < 8; k4++) {
      if (k4 < 7) ldfrag(k4 + 1, a0n, a1n, b0n, b1n);   // prefetch next frags
      acc[0][0] = wmma_f32(a0c, b0c, acc[0][0]);
      acc[0][1] = wmma_f32(a0c, b1c, acc[0][1]);
      acc[1][0] = wmma_f32(a1c, b0c, acc[1][0]);
      acc[1][1] = wmma_f32(a1c, b1c, acc[1][1]);
      a0c = a0n; a1c = a1n; b0c = b0n; b1c = b1n;
    }
    buf ^= 1;
  }

  // epilogue: C layout -> VGPR i: lanes0-15 row=i, lanes16-31 row=8+i
  const int rowHalf = lhi << 3;
#pragma unroll
  for (int mi = 0; mi < 2; mi++) {
#pragma unroll
    for (int ni = 0; ni < 2; ni++) {
      const int col = nBlock + wn * 32 + ni * 16 + l16;
      const float bcol = bias[col];
#pragma unroll
      for (int i = 0; i < 8; i++) {
        const int row = mBlock + wm * 32 + mi * 16 + rowHalf + i;
        float v = acc[mi][ni][i] + bcol;
        if (op == OP_GELU) {
          v = 0.5f * v * (1.0f + erff(v * 0.70710678118654752f));  // exact GELU
        } else if (op == OP_RES) {
          v += R[(size_t)row * Nn + col];
        }
        C[(size_t)row * Nn + col] = v;
      }
    }
  }
}

// ---------------------------------------------------------------------------
// Attention: one workgroup (4 waves) per (batch, head, 64-query tile).
// Per-head K^T (then V, paired layout) + the 64x512 score matrix live in
// 289 KB of the 320 KB WGP LDS -> no N^2 HBM traffic. Wave-parallel softmax
// (each wave owns the 16 rows it produced). Residual add fused.
// ---------------------------------------------------------------------------
#define SS         516   // score row stride (floats): 4r bank spread
#define KTP_STRIDE 528   // float2 row stride: +1 kpair => +32 banks
#define VP_STRIDE  80    // float2 row stride: +1 kpair => +32 banks

#define KV_FLOATS        (2 * (SEQ / 2) * VP_STRIDE)   // 40960 (>= Ktp 33792)
#define ATTN_SMEM_FLOATS (64 * SS + KV_FLOATS)         // 73984
#define ATTN_SMEM_BYTES  (ATTN_SMEM_FLOATS * 4)        // 295936 B

__global__ __launch_bounds__(128)
void attn_kernel(const float* __restrict__ Q, const float* __restrict__ Km,
                 const float* __restrict__ Vm, float* __restrict__ X) {
  extern __shared__ float smem[];
  float* S  = smem;             // 64 x SS scores
  float* KV = smem + 64 * SS;   // K^T pairs, later V pairs

  const int tid  = threadIdx.x;
  const int lane = tid & 31;
  const int wave = tid >> 5;            // 0..3, 16 query rows each
  const int l16  = lane & 15;
  const int lhi  = lane >> 4;
  const int khalf = lhi << 1;

  const int g  = blockIdx.x;
  const int qt = g & 7;
  const int h  = (g >> 3) % HEADS;
  const int b  = g / (8 * HEADS);

  const size_t headOff = (size_t)h * HDIM;
  const int qbase = qt * 64 + wave * 16;

  // preload Q fragments for this wave's 16 rows (16 k-steps of K=4)
  v2f aq[16];
  {
    const float* qrow = Q + ((size_t)(b * SEQ) + qbase + l16) * D_MODEL + headOff;
#pragma unroll
    for (int ks = 0; ks < 16; ks++)
      aq[ks] = *(const v2f*)(qrow + ks * 4 + khalf);
  }

  // stage K^T as (k,k+1) float2 pairs: Ktp[dp][n], dp=0..31, n=0..511
  {
    float2* Ktp = (float2*)KV;
    for (int it = 0; it < 128; it++) {
      int idx = it * 128 + tid;
      int n  = idx & 511;
      int dp = idx >> 9;
      float2 v = *(const float2*)(Km + ((size_t)(b * SEQ) + n) * D_MODEL +
                                  headOff + dp * 2);
      Ktp[dp * KTP_STRIDE + n] = v;
    }
  }
  __syncthreads();

  // S = (Q K^T) / sqrt(64); wave computes its own 16 rows vs all 512 keys
  {
    const float2* Ktp = (const float2*)KV;
    const int rbase = wave * 16 + (lhi << 3);
    for (int nt = 0; nt < 32; nt++) {
      v8f acc = {};
      v2f bc = *(const v2f*)&Ktp[lhi * KTP_STRIDE + nt * 16 + l16];
      v2f bn;
#pragma unroll
      for (int ks = 0; ks < 16; ks++) {
        if (ks < 15)
          bn = *(const v2f*)&Ktp[((ks + 1) * 2 + lhi) * KTP_STRIDE + nt * 16 + l16];
        acc = wmma_f32(aq[ks], bc, acc);
        bc = bn;
      }
#pragma unroll
      for (int i = 0; i < 8; i++)
        S[(rbase + i) * SS + nt * 16 + l16] = acc[i] * 0.125f;
    }
  }

  // wave-parallel softmax on this wave's 16 rows (intra-wave: no barrier)
  {
    for (int rr = 0; rr < 16; rr++) {
      float* srow = S + (size_t)(wave * 16 + rr) * SS;
      float v[16];
      float m = -1e30f;
#pragma unroll
      for (int j = 0; j < 16; j++) { v[j] = srow[lane + 32 * j]; m = fmaxf(m, v[j]); }
#pragma unroll
      for (int off = 16; off > 0; off >>= 1) m = fmaxf(m, __shfl_xor(m, off, 32));
      float s = 0.0f;
#pragma unroll
      for (int j = 0; j < 16; j++) { v[j] = expf(v[j] - m); s += v[j]; }
#pragma unroll
      for (int off = 16; off > 0; off >>= 1) s += __shfl_xor(s, off, 32);
      const float inv = 1.0f / s;
#pragma unroll
      for (int j = 0; j < 16; j++) srow[lane + 32 * j] = v[j] * inv;
    }
  }
  __syncthreads();   // all Ktp reads done -> safe to overwrite with V

  // stage V as (k,k+1) float2 pairs: Vp[kp][d], kp=0..255, d=0..63
  {
    float2* Vp = (float2*)KV;
    for (int it = 0; it < 32; it++) {
      int u  = it * 128 + tid;          // 4096 units
      int kp = u >> 4;
      int c4 = (u & 15) * 4;
      const float* vp = Vm + ((size_t)(b * SEQ) + 2 * kp) * D_MODEL + headOff + c4;
      float4 f0 = *(const float4*)vp;
      float4 f1 = *(const float4*)(vp + D_MODEL);
      float2* dst = &Vp[kp * VP_STRIDE + c4];
      dst[0] = make_float2(f0.x, f1.x);
      dst[1] = make_float2(f0.y, f1.y);
      dst[2] = make_float2(f0.z, f1.z);
      dst[3] = make_float2(f0.w, f1.w);
    }
  }
  __syncthreads();

  // O = P @ V ; fused residual: X += O
  {
    const float2* Vp = (const float2*)KV;
    v8f acc[4] = {};
    const float* Sr = S + ((size_t)(wave * 16 + l16)) * SS;

    v2f ac = *(const v2f*)(Sr + khalf);
    v2f bc[4], an, bn[4];
#pragma unroll
    for (int dt = 0; dt < 4; dt++)
      bc[dt] = *(const v2f*)&Vp[lhi * VP_STRIDE + dt * 16 + l16];

#pragma unroll 8
    for (int ks = 0; ks < 128; ks++) {
      if (ks < 127) {                       // prefetch next fragments
        an = *(const v2f*)(Sr + (ks + 1) * 4 + khalf);
        const int kp = (ks + 1) * 2 + lhi;
#pragma unroll
        for (int dt = 0; dt < 4; dt++)
          bn[dt] = *(const v2f*)&Vp[kp * VP_STRIDE + dt * 16 + l16];
      }
#pragma unroll
      for (int dt = 0; dt < 4; dt++)
        acc[dt] = wmma_f32(ac, bc[dt], acc[dt]);
      ac = an;
#pragma unroll
      for (int dt = 0; dt < 4; dt++) bc[dt] = bn[dt];
    }

    const int rbase = qt * 64 + wave * 16 + (lhi << 3);
#pragma unroll
    for (int dt = 0; dt < 4; dt++) {
      const int col = (int)headOff + dt * 16 + l16;
#pragma unroll
      for (int i = 0; i < 8; i++) {
        size_t off = ((size_t)(b * SEQ) + rbase + i) * D_MODEL + col;
        X[off] += acc[dt][i];
      }
    }
  }
}

// ---------------------------------------------------------------------------
// Orchestration
// ---------------------------------------------------------------------------
extern "C" void kernel_launch(void* const* d_in, const int* in_sizes, int n_in,
                              void* d_out, int out_size, void* d_ws, size_t ws_size,
                              hipStream_t stream) {
  const float* x  = (const float*)d_in[0];
  const float* Wq = (const float*)d_in[1];
  const float* bq = (const float*)d_in[2];
  const float* Wk = (const float*)d_in[3];
  const float* bk = (const float*)d_in[4];
  const float* Wv = (const float*)d_in[5];
  const float* bv = (const float*)d_in[6];
  const float* g1 = (const float*)d_in[7];
  const float* be1 = (const float*)d_in[8];
  const float* g2 = (const float*)d_in[9];
  const float* be2 = (const float*)d_in[10];
  const float* W0 = (const float*)d_in[11];
  const float* b0 = (const float*)d_in[12];
  const float* W1 = (const float*)d_in[13];
  const float* b1 = (const float*)d_in[14];

  const size_t actElems = (size_t)ROWS * D_MODEL;   // 1,572,864
  float* ws   = (float*)d_ws;
  float* xbuf = ws;
  float* Y    = ws + 1 * actElems;
  float* Qb   = ws + 2 * actElems;
  float* Kb   = ws + 3 * actElems;
  float* Vb   = ws + 4 * actElems;
  float* HH   = ws + 5 * actElems;                  // 2048 x 3072

  // never mutate inputs: work on a copy of x
  hipMemcpyAsync(xbuf, x, actElems * sizeof(float), hipMemcpyDeviceToDevice, stream);

  // attention kernel uses 289 KB of the 320 KB WGP LDS
  hipFuncSetAttribute((const void*)attn_kernel,
                      hipFuncAttributeMaxDynamicSharedMemorySize,
                      (int)ATTN_SMEM_BYTES);

  const dim3 gridD(ROWS / BM, D_MODEL / BN);   // 16 x 12
  const dim3 gridF(ROWS / BM, F_FF / BN);      // 16 x 48

  for (int blk = 0; blk < NBLOCKS; blk++) {
    ln_kernel<<<ROWS, 256, 0, stream>>>(xbuf, g1, be1, Y);

    gemm_kernel<<<gridD, 256, 0, stream>>>(Y, Wq, bq, Qb, nullptr, D_MODEL, D_MODEL, OP_NONE);
    gemm_kernel<<<gridD, 256, 0, stream>>>(Y, Wk, bk, Kb, nullptr, D_MODEL, D_MODEL, OP_NONE);
    gemm_kernel<<<gridD, 256, 0, stream>>>(Y, Wv, bv, Vb, nullptr, D_MODEL, D_MODEL, OP_NONE);

    attn_kernel<<<BATCH * HEADS * (SEQ / 64), 128, ATTN_SMEM_BYTES, stream>>>(Qb, Kb, Vb, xbuf);

    ln_kernel<<<ROWS, 256, 0, stream>>>(xbuf, g2, be2, Y);

    gemm_kernel<<<gridF, 256, 0, stream>>>(Y, W0, b0, HH, nullptr, D_MODEL, F_FF, OP_GELU);
    gemm_kernel<<<gridD, 256, 0, stream>>>(HH, W1, b1, xbuf, xbuf, F_FF, D_MODEL, OP_RES);
  }

  hipMemcpyAsync(d_out, xbuf, actElems * sizeof(float), hipMemcpyDeviceToDevice, stream);
}